// MyEGNNNet_21071109554393
// MI455X (gfx1250) — compile-verified
//
#include <hip/hip_runtime.h>

#define BB 2
#define NN 20000
#define EE 320000
#define CC 64
#define EPSV 1e-5f

typedef __attribute__((ext_vector_type(2))) float v2f;
typedef __attribute__((ext_vector_type(8))) float v8f;

__device__ __forceinline__ float sigmoidf(float x) {
    return 1.0f / (1.0f + __expf(-x));
}

// Fold Wk@Wa_k -> vk[64], Wq@Wa_q -> vq[64], We0@Wa_e -> se (scalar).
__global__ void prep_kernel(const float* __restrict__ Wk, const float* __restrict__ Wq,
                            const float* __restrict__ We, const float* __restrict__ Wa,
                            float* __restrict__ vk, float* __restrict__ vq,
                            float* __restrict__ se) {
    int k = threadIdx.x;  // 0..63
    float sk = 0.f, sq = 0.f;
    #pragma unroll 8
    for (int c = 0; c < CC; ++c) {
        sk += Wk[k * CC + c] * Wa[c];
        sq += Wq[k * CC + c] * Wa[CC + c];
    }
    vk[k] = sk;
    vq[k] = sq;
    if (k == 0) {
        float s = 0.f;
        for (int c = 0; c < CC; ++c) s += We[c] * Wa[2 * CC + c];
        *se = s;
    }
}

// xv = X @ value via V_WMMA_F32_16X16X4_F32.
// One wave computes a 16(M) x 64(N) tile; K=64 in steps of 4.
__global__ __launch_bounds__(256) void xv_kernel(const float* __restrict__ X,
                                                 const float* __restrict__ Wv,
                                                 float* __restrict__ xv) {
    const int lane  = threadIdx.x & 31;
    const int wtile = blockIdx.x * 8 + (threadIdx.x >> 5);
    if (wtile >= (BB * NN) / 16) return;          // wave-uniform: EXEC stays all-1s
    const int l    = lane & 15;
    const int m    = wtile * 16 + l;
    const int koff = (lane < 16) ? 0 : 2;         // A layout: lanes 0-15 K={0,1}, 16-31 K={2,3}
    v8f acc[4] = {};
    const float* arow = X + (long)m * CC;
    #pragma unroll 4
    for (int k0 = 0; k0 < CC; k0 += 4) {
        v2f a;
        a.x = arow[k0 + koff];
        a.y = arow[k0 + koff + 1];
        const float* b0 = Wv + (long)(k0 + koff) * CC;
        const float* b1 = Wv + (long)(k0 + koff + 1) * CC;
        #pragma unroll
        for (int t = 0; t < 4; ++t) {
            v2f bf;
            bf.x = b0[t * 16 + l];
            bf.y = b1[t * 16 + l];
            acc[t] = __builtin_amdgcn_wmma_f32_16x16x4_f32(
                false, a, false, bf, (short)0, acc[t], false, false);
        }
    }
    // C/D layout: VGPR r -> row r (lanes 0-15) / row r+8 (lanes 16-31), col = lane%16
    const int rowb = wtile * 16 + ((lane < 16) ? 0 : 8);
    #pragma unroll
    for (int r = 0; r < 8; ++r) {
        float* o = xv + (long)(rowb + r) * CC + l;
        o[0]  = acc[0][r];
        o[16] = acc[1][r];
        o[32] = acc[2][r];
        o[48] = acc[3][r];
    }
}

// Edge phase: 2 edges per wave, 16 lanes per edge, float4 (4 channels) per lane.
// att = sigmoid(dot(xv[dst],vk) + dot(xv[src],vq) + ew*se + ba)
// aggr[dst] += att * xv[src] * sigmoid(ew * We0)   (f32 global atomics, L2-resident)
__global__ __launch_bounds__(256) void edge_kernel(const float* __restrict__ xv,
                                                   const int* __restrict__ ei,
                                                   const float* __restrict__ ew,
                                                   const float* __restrict__ vk,
                                                   const float* __restrict__ vq,
                                                   const float* __restrict__ sep,
                                                   const float* __restrict__ ba,
                                                   const float* __restrict__ We,
                                                   float* __restrict__ aggr) {
    const int lane = threadIdx.x & 31;
    const int half = lane >> 4;
    const int l    = lane & 15;
    const long wid  = (long)blockIdx.x * 8 + (threadIdx.x >> 5);
    const long eidx = wid * 2 + half;             // in [0, B*E), grid sized exactly

    const int  b = (eidx >= EE) ? 1 : 0;
    const int  e = (int)(eidx - (long)b * EE);
    const int* src = ei;
    const int* dst = ei + EE;
    const int  s = src[e];
    const int  d = dst[e];
    const float w = ew[e];

    const float4* xs = (const float4*)(xv + ((long)b * NN + s) * CC);
    const float4* xd = (const float4*)(xv + ((long)b * NN + d) * CC);
    const float4 vs = xs[l];
    const float4 vd = xd[l];
    const float4 fk = ((const float4*)vk)[l];
    const float4 fq = ((const float4*)vq)[l];

    float part = vd.x * fk.x + vd.y * fk.y + vd.z * fk.z + vd.w * fk.w
               + vs.x * fq.x + vs.y * fq.y + vs.z * fq.z + vs.w * fq.w;
    // xor masks 1..8 stay inside each 16-lane half on wave32
    part += __shfl_xor(part, 1, 32);
    part += __shfl_xor(part, 2, 32);
    part += __shfl_xor(part, 4, 32);
    part += __shfl_xor(part, 8, 32);

    const float att = sigmoidf(part + w * sep[0] + ba[0]);
    const float4 we4 = ((const float4*)We)[l];
    float mx = att * vs.x * sigmoidf(w * we4.x);
    float my = att * vs.y * sigmoidf(w * we4.y);
    float mz = att * vs.z * sigmoidf(w * we4.z);
    float mw = att * vs.w * sigmoidf(w * we4.w);

    float* ap = aggr + ((long)b * NN + d) * CC + l * 4;
    atomicAdd(ap + 0, mx);
    atomicAdd(ap + 1, my);
    atomicAdd(ap + 2, mz);
    atomicAdd(ap + 3, mw);
}

// h = [xv | aggr] @ Wc + bc (K=128 WMMA chain), then LayerNorm + ReLU + residual.
__global__ __launch_bounds__(256) void fuse_kernel(const float* __restrict__ xv,
                                                   const float* __restrict__ aggr,
                                                   const float* __restrict__ Wc,
                                                   const float* __restrict__ bc,
                                                   const float* __restrict__ gamma,
                                                   const float* __restrict__ beta,
                                                   float* __restrict__ out) {
    const int lane  = threadIdx.x & 31;
    const int wtile = blockIdx.x * 8 + (threadIdx.x >> 5);
    if (wtile >= (BB * NN) / 16) return;          // wave-uniform
    const int l    = lane & 15;
    const int m    = wtile * 16 + l;
    const int koff = (lane < 16) ? 0 : 2;
    v8f acc[4] = {};
    const float* arow0 = xv + (long)m * CC;
    const float* arow1 = aggr + (long)m * CC - CC;  // so ar[k] indexes aggr[m][k-64]
    #pragma unroll 4
    for (int k0 = 0; k0 < 2 * CC; k0 += 4) {
        const float* ar = (k0 < CC) ? arow0 : arow1;
        v2f a;
        a.x = ar[k0 + koff];
        a.y = ar[k0 + koff + 1];
        const float* b0 = Wc + (long)(k0 + koff) * CC;
        const float* b1 = Wc + (long)(k0 + koff + 1) * CC;
        #pragma unroll
        for (int t = 0; t < 4; ++t) {
            v2f bf;
            bf.x = b0[t * 16 + l];
            bf.y = b1[t * 16 + l];
            acc[t] = __builtin_amdgcn_wmma_f32_16x16x4_f32(
                false, a, false, bf, (short)0, acc[t], false, false);
        }
    }

    float bcv[4], gv[4], bev[4];
    #pragma unroll
    for (int t = 0; t < 4; ++t) {
        bcv[t] = bc[t * 16 + l];
        gv[t]  = gamma[t * 16 + l];
        bev[t] = beta[t * 16 + l];
    }

    const int rbase = (lane < 16) ? 0 : 8;
    #pragma unroll
    for (int r = 0; r < 8; ++r) {
        float h0 = acc[0][r] + bcv[0];
        float h1 = acc[1][r] + bcv[1];
        float h2 = acc[2][r] + bcv[2];
        float h3 = acc[3][r] + bcv[3];
        float s = h0 + h1 + h2 + h3;
        float q = h0 * h0 + h1 * h1 + h2 * h2 + h3 * h3;
        // each row's 64 values live in one 16-lane half; reduce within it
        s += __shfl_xor(s, 1, 32); q += __shfl_xor(q, 1, 32);
        s += __shfl_xor(s, 2, 32); q += __shfl_xor(q, 2, 32);
        s += __shfl_xor(s, 4, 32); q += __shfl_xor(q, 4, 32);
        s += __shfl_xor(s, 8, 32); q += __shfl_xor(q, 8, 32);
        const float mean = s * (1.0f / 64.0f);
        const float var  = q * (1.0f / 64.0f) - mean * mean;
        const float inv  = rsqrtf(var + EPSV);

        const int row = wtile * 16 + rbase + r;
        const float* xr = xv + (long)row * CC;
        float* o = out + (long)row * CC;
        h0 = (h0 - mean) * inv * gv[0] + bev[0];
        h1 = (h1 - mean) * inv * gv[1] + bev[1];
        h2 = (h2 - mean) * inv * gv[2] + bev[2];
        h3 = (h3 - mean) * inv * gv[3] + bev[3];
        o[l]      = xr[l]      + fmaxf(h0, 0.0f);
        o[l + 16] = xr[l + 16] + fmaxf(h1, 0.0f);
        o[l + 32] = xr[l + 32] + fmaxf(h2, 0.0f);
        o[l + 48] = xr[l + 48] + fmaxf(h3, 0.0f);
    }
}

extern "C" void kernel_launch(void* const* d_in, const int* in_sizes, int n_in,
                              void* d_out, int out_size, void* d_ws, size_t ws_size,
                              hipStream_t stream) {
    const float* X     = (const float*)d_in[0];
    const int*   ei    = (const int*)d_in[1];
    const float* ew    = (const float*)d_in[2];
    const float* Wv    = (const float*)d_in[3];
    const float* Wk    = (const float*)d_in[4];
    const float* Wq    = (const float*)d_in[5];
    const float* We    = (const float*)d_in[6];
    const float* Wa    = (const float*)d_in[7];
    const float* ba    = (const float*)d_in[8];
    const float* Wc    = (const float*)d_in[9];
    const float* bc    = (const float*)d_in[10];
    const float* gamma = (const float*)d_in[11];
    const float* beta  = (const float*)d_in[12];
    float* out = (float*)d_out;

    const size_t nElem = (size_t)BB * NN * CC;   // 2,560,000 floats
    float* xv   = (float*)d_ws;
    float* aggr = xv + nElem;
    float* vk   = aggr + nElem;
    float* vq   = vk + CC;
    float* se   = vq + CC;

    hipMemsetAsync(aggr, 0, nElem * sizeof(float), stream);
    prep_kernel<<<1, 64, 0, stream>>>(Wk, Wq, We, Wa, vk, vq, se);

    const int rowTiles = (BB * NN) / 16;         // 2500
    const int gemmBlocks = (rowTiles + 7) / 8;   // 8 waves per 256-thread block
    xv_kernel<<<gemmBlocks, 256, 0, stream>>>(X, Wv, xv);

    // B*E = 640000 edge-instances, 2 per wave, 8 waves per block -> exactly 40000 blocks
    edge_kernel<<<(BB * EE) / 16, 256, 0, stream>>>(xv, ei, ew, vk, vq, se, ba, We, aggr);

    fuse_kernel<<<gemmBlocks, 256, 0, stream>>>(xv, aggr, Wc, bc, gamma, beta, out);
}